// QuanvolutionHybrid_65481071395558
// MI455X (gfx1250) — compile-verified
//
#include <hip/hip_runtime.h>
#include <hip/hip_bf16.h>
#include <math.h>

typedef __attribute__((ext_vector_type(2))) float v2f;
typedef __attribute__((ext_vector_type(8))) float v8f;

// ---------------------------------------------------------------------------
// Kernel 1: quantum features. One thread per 2x2 patch; 4-qubit state (16
// complex amps) lives entirely in registers; all gates unrolled.
// Output layout: fbuf[b*784 + patch*4 + q]  (== f[b][c][h][w] with
// flat j = c*196+h*14+w, matching the reference's bug-compatible reshape).
// ---------------------------------------------------------------------------
__device__ __forceinline__ void gate_ry(float re[16], float im[16], int stride,
                                        float c, float s) {
#pragma unroll
  for (int i = 0; i < 16; ++i) {
    if (i & stride) continue;
    int j = i | stride;
    float r0 = re[i], r1 = re[j], i0 = im[i], i1 = im[j];
    re[i] = c * r0 - s * r1;  im[i] = c * i0 - s * i1;
    re[j] = s * r0 + c * r1;  im[j] = s * i0 + c * i1;
  }
}

__device__ __forceinline__ void gate_rx(float re[16], float im[16], int stride,
                                        float c, float s) {
#pragma unroll
  for (int i = 0; i < 16; ++i) {
    if (i & stride) continue;
    int j = i | stride;
    float r0 = re[i], r1 = re[j], i0 = im[i], i1 = im[j];
    // new0 = c*v0 - i*s*v1 ; new1 = c*v1 - i*s*v0
    re[i] = c * r0 + s * i1;  im[i] = c * i0 - s * r1;
    re[j] = c * r1 + s * i0;  im[j] = c * i1 - s * r0;
  }
}

__device__ __forceinline__ void gate_rz(float re[16], float im[16], int stride,
                                        float c, float s) {
#pragma unroll
  for (int i = 0; i < 16; ++i) {
    float r = re[i], q = im[i];
    if (i & stride) {           // phase e^{+i t/2}
      re[i] = c * r - s * q;  im[i] = c * q + s * r;
    } else {                    // phase e^{-i t/2}
      re[i] = c * r + s * q;  im[i] = c * q - s * r;
    }
  }
}

__device__ __forceinline__ void gate_cnot(float re[16], float im[16],
                                          int cstride, int tstride) {
#pragma unroll
  for (int i = 0; i < 16; ++i) {
    if (!(i & cstride)) continue;  // control bit must be 1
    if (i & tstride) continue;     // visit target=0 element of the pair
    int j = i | tstride;
    float tr = re[i]; re[i] = re[j]; re[j] = tr;
    float ti = im[i]; im[i] = im[j]; im[j] = ti;
  }
}

__global__ void quantum_kernel(const float* __restrict__ x,
                               const float* __restrict__ qw,
                               float* __restrict__ fbuf, int B) {
  int n = blockIdx.x * blockDim.x + threadIdx.x;
  int total = B * 196;
  if (n >= total) return;
  int b  = n / 196;
  int pt = n - b * 196;
  int pr = pt / 14, pc = pt - pr * 14;
  const float* img = x + (size_t)b * 784;
  int r0 = pr * 2, c0 = pc * 2;
  float px[4];
  px[0] = img[r0 * 28 + c0];
  px[1] = img[r0 * 28 + c0 + 1];
  px[2] = img[(r0 + 1) * 28 + c0];
  px[3] = img[(r0 + 1) * 28 + c0 + 1];

  float ca[4], sa[4];
#pragma unroll
  for (int k = 0; k < 4; ++k) sincosf(px[k] * 0.5f, &sa[k], &ca[k]);

  float gc[5], gs[5];
#pragma unroll
  for (int k = 0; k < 5; ++k) sincosf(qw[k] * 0.5f, &gs[k], &gc[k]);

  float re[16], im[16];
#pragma unroll
  for (int i = 0; i < 16; ++i) {
    int a = (i >> 3) & 1, bq = (i >> 2) & 1, cq = (i >> 1) & 1, d = i & 1;
    re[i] = (a ? sa[0] : ca[0]) * (bq ? sa[1] : ca[1]) *
            (cq ? sa[2] : ca[2]) * (d ? sa[3] : ca[3]);
    im[i] = 0.f;
  }
  // qubit q <-> bit stride (8 >> q)
  gate_ry(re, im, 8, gc[0], gs[0]);   // RY(q0)
  gate_rx(re, im, 4, gc[1], gs[1]);   // RX(q1)
  gate_cnot(re, im, 8, 4);            // CNOT(0,1)
  gate_rz(re, im, 2, gc[2], gs[2]);   // RZ(q2)
  gate_cnot(re, im, 2, 1);            // CNOT(2,3)
  gate_rx(re, im, 1, gc[3], gs[3]);   // RX(q3)
  gate_cnot(re, im, 4, 2);            // CNOT(1,2)
  gate_ry(re, im, 8, gc[4], gs[4]);   // RY(q0)

  float m[4] = {0.f, 0.f, 0.f, 0.f};
#pragma unroll
  for (int i = 0; i < 16; ++i) {
    float p = re[i] * re[i] + im[i] * im[i];
#pragma unroll
    for (int q = 0; q < 4; ++q) m[q] += (i & (8 >> q)) ? -p : p;
  }
  float* o = fbuf + (size_t)b * 784 + pt * 4;
#pragma unroll
  for (int q = 0; q < 4; ++q) o[q] = m[q];
}

// ---------------------------------------------------------------------------
// Kernel 2/4: 3x3 SAME conv on (B,4,14,14). doBn=1 applies bn1+relu to input
// taps on the fly (bn is elementwise, so no extra buffer needed).
// ---------------------------------------------------------------------------
__global__ void conv3x3_kernel(const float* __restrict__ in,
                               const float* __restrict__ w,
                               const float* __restrict__ stats,   // mean[4], inv[4]
                               const float* __restrict__ gamma,
                               const float* __restrict__ beta,
                               float* __restrict__ out, int B, int doBn) {
  int idx = blockIdx.x * blockDim.x + threadIdx.x;
  int total = B * 784;
  if (idx >= total) return;
  int b   = idx / 784;
  int rem = idx - b * 784;
  int co  = rem / 196;
  int s   = rem - co * 196;
  int h = s / 14, wc = s - h * 14;
  float acc = 0.f;
#pragma unroll
  for (int ci = 0; ci < 4; ++ci) {
    float mm = 0.f, inv = 1.f, gg = 1.f, bb = 0.f;
    if (doBn) { mm = stats[ci]; inv = stats[4 + ci]; gg = gamma[ci]; bb = beta[ci]; }
    const float* inc = in + (size_t)b * 784 + ci * 196;
    const float* wk  = w + (co * 4 + ci) * 9;
#pragma unroll
    for (int kh = 0; kh < 3; ++kh) {
      int ih = h + kh - 1;
      if (ih < 0 || ih >= 14) continue;
#pragma unroll
      for (int kw = 0; kw < 3; ++kw) {
        int iw = wc + kw - 1;
        if (iw < 0 || iw >= 14) continue;
        float v = inc[ih * 14 + iw];
        if (doBn) v = fmaxf(fmaf(v - mm, inv * gg, bb), 0.f);
        acc = fmaf(wk[kh * 3 + kw], v, acc);
      }
    }
  }
  out[idx] = acc;
}

// ---------------------------------------------------------------------------
// Kernel 3/5: per-channel batch stats (mean, rsqrt(var+eps)). One block per
// channel; deterministic LDS tree reduction (no atomics).
// ---------------------------------------------------------------------------
__global__ void bn_stats_kernel(const float* __restrict__ src,
                                float* __restrict__ statsOut, int B) {
  int c = blockIdx.x;
  int t = threadIdx.x;
  int N = B * 196;
  float sum = 0.f, sq = 0.f;
  for (int i = t; i < N; i += blockDim.x) {
    int b = i / 196, s = i - b * 196;
    float v = src[(size_t)b * 784 + c * 196 + s];
    sum += v; sq += v * v;
  }
  __shared__ float s1[256], s2[256];
  s1[t] = sum; s2[t] = sq;
  __syncthreads();
  for (int off = 128; off > 0; off >>= 1) {
    if (t < off) { s1[t] += s1[t + off]; s2[t] += s2[t + off]; }
    __syncthreads();
  }
  if (t == 0) {
    float invN = 1.f / (float)N;
    float m = s1[0] * invN;
    float var = fmaxf(s2[0] * invN - m * m, 0.f);
    statsOut[c]     = m;
    statsOut[4 + c] = rsqrtf(var + 1e-5f);
  }
}

// ---------------------------------------------------------------------------
// Kernel 6: fused bn2 + residual + relu + FC (WMMA f32 16x16x4) + log_softmax.
// One wave per 16-row batch tile; N=16 cols (0..9 = fc rows, rest zero);
// K=784 stepped by 4.
// A frag (16x4 f32, ISA 7.12.2): lanes 0-15 -> M=lane, K={k0,k0+1};
//                                lanes 16-31 -> M=lane-16, K={k0+2,k0+3}.
// B frag (4x16 f32, mirrored):   lanes 0-15 -> N=lane, K={k0,k0+1};
//                                lanes 16-31 -> N=lane-16, K={k0+2,k0+3}.
// D (16x16 f32): VGPR r: lanes 0-15 -> M=r, lanes 16-31 -> M=r+8; N=lane&15.
// ---------------------------------------------------------------------------
__global__ void fc_softmax_wmma_kernel(const float* __restrict__ y2,
                                       const float* __restrict__ f,
                                       const float* __restrict__ stats2,
                                       const float* __restrict__ g2,
                                       const float* __restrict__ be2,
                                       const float* __restrict__ fcw,
                                       const float* __restrict__ fcb,
                                       float* __restrict__ out, int B) {
  __shared__ float lds[8][16][10];
  int wv   = threadIdx.x >> 5;
  int lane = threadIdx.x & 31;
  int n_   = lane & 15;
  int hi   = lane >> 4;
  int nTiles = B >> 4;
  int tile = blockIdx.x * 8 + wv;
  bool active = tile < nTiles;
  int rowBase = tile * 16;

  if (active) {
    int arow = rowBase + n_;
    const float* y2row = y2 + (size_t)arow * 784;
    const float* frow  = f  + (size_t)arow * 784;
    v8f acc = {};
    for (int k0 = 0; k0 < 784; k0 += 4) {
      int c = k0 / 196;                       // wave-uniform
      float mm = stats2[c], inv = stats2[4 + c];
      float sc = inv * g2[c], bb = be2[c];
      int ka = k0 + 2 * hi;
      float a0 = fmaxf(fmaf(y2row[ka]     - mm, sc, bb) + frow[ka],     0.f);
      float a1 = fmaxf(fmaf(y2row[ka + 1] - mm, sc, bb) + frow[ka + 1], 0.f);
      v2f av = {a0, a1};
      float b0 = 0.f, b1 = 0.f;
      if (n_ < 10) { b0 = fcw[n_ * 784 + ka]; b1 = fcw[n_ * 784 + ka + 1]; }
      v2f bv = {b0, b1};
      acc = __builtin_amdgcn_wmma_f32_16x16x4_f32(
          /*neg_a=*/false, av, /*neg_b=*/false, bv,
          /*c_mod=*/(short)0, acc, /*reuse_a=*/false, /*reuse_b=*/false);
    }
    if (n_ < 10) {
      float fb = fcb[n_];
#pragma unroll
      for (int r = 0; r < 8; ++r) {
        int M = r + 8 * hi;
        lds[wv][M][n_] = acc[r] + fb;
      }
    }
  }
  __syncthreads();
  if (active && hi == 0) {
    int M = n_;
    float l[10];
    float mx = -1e30f;
#pragma unroll
    for (int q = 0; q < 10; ++q) { l[q] = lds[wv][M][q]; mx = fmaxf(mx, l[q]); }
    float s = 0.f;
#pragma unroll
    for (int q = 0; q < 10; ++q) s += expf(l[q] - mx);
    float lse = logf(s) + mx;
    float* o = out + (size_t)(rowBase + M) * 10;
#pragma unroll
    for (int q = 0; q < 10; ++q) o[q] = l[q] - lse;
  }
}

// ---------------------------------------------------------------------------
// Host launcher
// ---------------------------------------------------------------------------
extern "C" void kernel_launch(void* const* d_in, const int* in_sizes, int n_in,
                              void* d_out, int out_size, void* d_ws, size_t ws_size,
                              hipStream_t stream) {
  const float* x      = (const float*)d_in[0];
  const float* qw     = (const float*)d_in[1];
  const float* w1     = (const float*)d_in[2];
  const float* g1     = (const float*)d_in[3];
  const float* b1     = (const float*)d_in[4];
  const float* w2     = (const float*)d_in[5];
  const float* g2     = (const float*)d_in[6];
  const float* b2     = (const float*)d_in[7];
  const float* fcw    = (const float*)d_in[8];
  const float* fcb    = (const float*)d_in[9];
  float* out = (float*)d_out;

  int B = in_sizes[0] / 784;
  size_t E = (size_t)B * 784;

  float* fbuf   = (float*)d_ws;            // B*784
  float* y1     = fbuf + E;                // B*784
  float* y2     = y1 + E;                  // B*784
  float* stats1 = y2 + E;                  // 8 floats
  float* stats2 = stats1 + 8;              // 8 floats

  int threads = 256;
  int gQ = (B * 196 + threads - 1) / threads;
  quantum_kernel<<<gQ, threads, 0, stream>>>(x, qw, fbuf, B);

  int gC = (B * 784 + threads - 1) / threads;
  conv3x3_kernel<<<gC, threads, 0, stream>>>(fbuf, w1, nullptr, nullptr, nullptr,
                                             y1, B, 0);
  bn_stats_kernel<<<4, 256, 0, stream>>>(y1, stats1, B);
  conv3x3_kernel<<<gC, threads, 0, stream>>>(y1, w2, stats1, g1, b1, y2, B, 1);
  bn_stats_kernel<<<4, 256, 0, stream>>>(y2, stats2, B);

  int nTiles = B / 16;
  int gF = (nTiles + 7) / 8;
  fc_softmax_wmma_kernel<<<gF, 256, 0, stream>>>(y2, fbuf, stats2, g2, b2,
                                                 fcw, fcb, out, B);
}